// ProposalLayer_59940563583605
// MI455X (gfx1250) — compile-verified
//
#include <hip/hip_runtime.h>
#include <stdint.h>

// ---------------------------------------------------------------------------
// ProposalLayer (RPN): H=W=480, K=9 anchors, stride 4, NMS_PRE=3000,
// NMS_POST=300, IoU thr=0.6, image 1920.
// cls_scores: (1,18,480,480) f32   bbox_deltas: (1,36,480,480) f32
// out: (1,300,5) f32 = 1500 floats
// ---------------------------------------------------------------------------

#define HW      230400      // 480*480
#define NTOT    2073600     // 9*HW
#define NF4     518400      // NTOT/4
#define PLANE4  57600       // HW/4
#define NPRE    3000
#define NPOST   300
#define CAP     16384       // candidate capacity after radix-bin threshold
#define MROW    128         // mask row stride in words (94 used, padded)
#define RING    32          // mask rows in flight in LDS ring

__device__ __forceinline__ uint32_t fkey(float f) {
    uint32_t b = __float_as_uint(f);
    return b ^ ((b & 0x80000000u) ? 0xFFFFFFFFu : 0x80000000u);
}
__device__ __forceinline__ float key2f(uint32_t k) {
    return __uint_as_float(k ^ ((k & 0x80000000u) ? 0x80000000u : 0xFFFFFFFFu));
}
__device__ __forceinline__ uint32_t lds_off(const void* p) {
    // generic pointer to __shared__ has the WG-relative LDS offset in its low 32 bits
    return (uint32_t)(uintptr_t)p;
}

// ---- K0: init workspace + output -------------------------------------------
__global__ __launch_bounds__(256) void k0_init(uint32_t* hist, uint32_t* counter,
                                               float* sscore, uint32_t* ssidx,
                                               float* out) {
    int g = blockIdx.x * 256 + threadIdx.x;
    if (g < 4096) hist[g] = 0u;
    if (g < NPRE) { sscore[g] = -3.4028235e38f; ssidx[g] = 0u; }
    if (g < NPOST * 5) out[g] = 0.0f;
    if (g == 0) *counter = 0u;
}

// ---- K1: histogram of 12-bit key prefixes ----------------------------------
__global__ __launch_bounds__(256) void k1_hist(const float4* __restrict__ cls4,
                                               uint32_t* __restrict__ hist) {
    __shared__ uint32_t lh[4096];
    for (int b = threadIdx.x; b < 4096; b += 256) lh[b] = 0u;
    __syncthreads();
    const int stride = gridDim.x * 256;
    for (int i4 = blockIdx.x * 256 + threadIdx.x; i4 < NF4; i4 += stride) {
        int k = i4 / PLANE4;                      // which of the 9 even channels
        const float4* p = cls4 + (i4 + k * PLANE4);
        __builtin_prefetch((const void*)(p + stride), 0, 0);  // global_prefetch_b8
        float4 v = *p;
        atomicAdd(&lh[fkey(v.x) >> 20], 1u);
        atomicAdd(&lh[fkey(v.y) >> 20], 1u);
        atomicAdd(&lh[fkey(v.z) >> 20], 1u);
        atomicAdd(&lh[fkey(v.w) >> 20], 1u);
    }
    __syncthreads();
    for (int b = threadIdx.x; b < 4096; b += 256) {
        uint32_t c = lh[b];
        if (c) atomicAdd(&hist[b], c);
    }
}

// ---- K2: find threshold bin (cumulative from top >= NPRE) ------------------
__global__ void k2_thresh(const uint32_t* __restrict__ hist, int* __restrict__ binB) {
    if (threadIdx.x == 0 && blockIdx.x == 0) {
        int cum = 0, B = 0;
        for (int b = 4095; b >= 0; --b) {
            cum += (int)hist[b];
            if (cum >= NPRE) { B = b; break; }
        }
        *binB = B;
    }
}

// ---- K3: compact all elements with bin >= B --------------------------------
__global__ __launch_bounds__(256) void k3_compact(const float4* __restrict__ cls4,
                                                  const int* __restrict__ binB,
                                                  uint32_t* __restrict__ counter,
                                                  uint32_t* __restrict__ ckey,
                                                  uint32_t* __restrict__ cidx) {
    const int B = *binB;
    const int stride = gridDim.x * 256;
    for (int i4 = blockIdx.x * 256 + threadIdx.x; i4 < NF4; i4 += stride) {
        int k = i4 / PLANE4;
        const float4* p = cls4 + (i4 + k * PLANE4);
        __builtin_prefetch((const void*)(p + stride), 0, 0);
        float4 v = *p;
        uint32_t i0 = (uint32_t)i4 * 4u;
        float vv[4] = {v.x, v.y, v.z, v.w};
#pragma unroll
        for (int c = 0; c < 4; ++c) {
            uint32_t key = fkey(vv[c]);
            if ((int)(key >> 20) >= B) {
                uint32_t pos = atomicAdd(counter, 1u);
                if (pos < CAP) { ckey[pos] = key; cidx[pos] = i0 + (uint32_t)c; }
            }
        }
    }
}

// ---- K4: exact rank-sort of candidates (key desc, index asc) ---------------
__global__ __launch_bounds__(256) void k4_rank(const uint32_t* __restrict__ ckey,
                                               const uint32_t* __restrict__ cidx,
                                               const uint32_t* __restrict__ counter,
                                               float* __restrict__ sscore,
                                               uint32_t* __restrict__ ssidx) {
    __shared__ uint32_t tk[1024];
    __shared__ uint32_t ti[1024];
    const int j = blockIdx.x * 256 + threadIdx.x;
    const int n = (int)min(*counter, (uint32_t)CAP);
    uint32_t kj = 0u, ij = 0u;
    const bool act = (j < n);
    if (act) { kj = ckey[j]; ij = cidx[j]; }
    int rank = 0;
    for (int t0 = 0; t0 < n; t0 += 1024) {
        const int cnt = min(1024, n - t0);
        __syncthreads();
        for (int s = threadIdx.x; s < cnt; s += 256) { tk[s] = ckey[t0 + s]; ti[s] = cidx[t0 + s]; }
        __syncthreads();
        if (act) {
            for (int s = 0; s < cnt; ++s) {
                uint32_t kt = tk[s];
                rank += (kt > kj) || (kt == kj && ti[s] < ij);
            }
        }
    }
    if (act && rank < NPRE) { sscore[rank] = key2f(kj); ssidx[rank] = ij; }
}

// ---- K5: decode boxes through the reference's reshape-gather ---------------
__global__ __launch_bounds__(256) void k5_boxes(const uint32_t* __restrict__ ssidx,
                                                const float* __restrict__ deltas,
                                                float4* __restrict__ boxes4) {
    int t = blockIdx.x * 256 + threadIdx.x;
    if (t >= NPRE) return;
    uint32_t i = ssidx[t];
    uint32_t k2 = i % 9u;
    uint32_t q  = i / 9u;
    uint32_t w2 = q % 480u;
    uint32_t h2 = q / 480u;
    float bc[4];
#pragma unroll
    for (uint32_t c = 0; c < 4; ++c) {
        uint32_t f  = (k2 * 4u + c) * (uint32_t)HW + h2 * 480u + w2;
        uint32_t hh = f / 17280u;           // regions strides (17280,36,4,1)
        uint32_t r1 = f - hh * 17280u;
        uint32_t ww = r1 / 36u;
        uint32_t r2 = r1 - ww * 36u;
        uint32_t kk = r2 >> 2;
        uint32_t cc = r2 & 3u;
        uint32_t rg = kk / 3u, sg = kk - rg * 3u;
        float rr = (rg == 0u) ? 0.5f : (rg == 1u) ? 1.0f : 2.0f;
        float ss = (sg == 0u) ? 128.0f : (sg == 1u) ? 256.0f : 512.0f;
        bool isX   = (cc == 0u || cc == 2u);
        float cv   = isX ? (2.0f * (float)hh + 2.0f) : (2.0f * (float)ww + 2.0f);
        float half = isX ? (ss * 0.5f) : (ss * rr * 0.5f);
        float a    = (cc < 2u) ? (cv - half) : (cv + half);
        float d    = deltas[(4u * kk + cc) * (uint32_t)HW + hh * 480u + ww];
        bc[c] = fminf(fmaxf(a + d, 0.0f), 1920.0f);
    }
    boxes4[t] = make_float4(bc[0], bc[1], bc[2], bc[3]);
}

// ---- K6: pairwise IoU suppression bitmask (3000 rows x 94 words) -----------
__global__ __launch_bounds__(256) void k6_mask(const float4* __restrict__ boxes4,
                                               uint32_t* __restrict__ mask) {
    int gid = blockIdx.x * 256 + threadIdx.x;
    if (gid >= NPRE * 94) return;
    int i = gid / 94, w = gid - i * 94;
    float4 bi = boxes4[i];
    float areai = (bi.z - bi.x) * (bi.w - bi.y);
    uint32_t bits = 0u;
    int j0 = w * 32;
#pragma unroll 4
    for (int b = 0; b < 32; ++b) {
        int j = j0 + b;
        if (j >= NPRE) break;
        float4 bj = boxes4[j];
        float areaj = (bj.z - bj.x) * (bj.w - bj.y);
        float ix1 = fmaxf(bi.x, bj.x), iy1 = fmaxf(bi.y, bj.y);
        float ix2 = fminf(bi.z, bj.z), iy2 = fminf(bi.w, bj.w);
        float inter = fmaxf(ix2 - ix1, 0.0f) * fmaxf(iy2 - iy1, 0.0f);
        float iou = inter / (areai + areaj - inter + 1e-9f);
        if (iou > 0.6f) bits |= (1u << b);
    }
    mask[(size_t)i * MROW + w] = bits;
}

// ---- K7: serial greedy NMS scan, single wave32 -----------------------------
// Mask rows streamed through a 32-row LDS ring with async global->LDS loads.
// ASYNCcnt completes in order, so "s_wait_asynccnt RING-1" guarantees the
// oldest outstanding row (row i) has fully landed in LDS.
__global__ __launch_bounds__(32) void k7_scan(const uint32_t* __restrict__ mask,
                                              const float* __restrict__ sscore,
                                              const float4* __restrict__ boxes4,
                                              float* __restrict__ out) {
    __shared__ __align__(16) uint32_t ring[RING * MROW];  // 16 KB
    __shared__ uint32_t remv[96];
    const int lane = threadIdx.x;
    remv[lane] = 0u; remv[lane + 32] = 0u; remv[lane + 64] = 0u;
    __syncthreads();  // single-wave WG: lowers to s_nop + waits

    const uint32_t ldsBase = lds_off(&ring[0]);
    // prologue: fill the ring (one async b128 per lane = one 512B row per op)
    for (int r = 0; r < RING; ++r) {
        unsigned long long g = (unsigned long long)(uintptr_t)(mask + (size_t)r * MROW)
                               + (unsigned long long)(lane * 16);
        uint32_t l = ldsBase + (uint32_t)r * (MROW * 4u) + (uint32_t)lane * 16u;
        asm volatile("global_load_async_to_lds_b128 %0, %1, off"
                     :: "v"(l), "v"(g) : "memory");
    }

    int cnt = 0;
    for (int i = 0; i < NPRE; ++i) {
        asm volatile("s_wait_asynccnt 31" ::: "memory");  // row i is resident
        const int slot = i & (RING - 1);
        uint32_t wv = remv[i >> 5];                       // uniform LDS broadcast
        bool keep = ((wv >> (i & 31)) & 1u) == 0u;        // uniform across wave
        if (keep) {
            if (cnt < NPOST) {
                if (lane == 0) {
                    out[cnt * 5 + 0] = sscore[i];
                } else if (lane <= 4) {
                    const float* bp = (const float*)(boxes4 + i);
                    out[cnt * 5 + lane] = bp[lane - 1];
                }
            }
            cnt++;
            const uint32_t base = (uint32_t)slot * MROW;
            remv[lane]      |= ring[base + lane];
            remv[lane + 32] |= ring[base + lane + 32];
            remv[lane + 64] |= ring[base + lane + 64];
            __syncthreads();  // order remv writes before next iteration's read
        }
        const int nr = i + RING;
        if (nr < NPRE) {  // refill the slot we just retired
            unsigned long long g = (unsigned long long)(uintptr_t)(mask + (size_t)nr * MROW)
                                   + (unsigned long long)(lane * 16);
            uint32_t l = ldsBase + (uint32_t)slot * (MROW * 4u) + (uint32_t)lane * 16u;
            asm volatile("global_load_async_to_lds_b128 %0, %1, off"
                         :: "v"(l), "v"(g) : "memory");
        }
    }
    asm volatile("s_wait_asynccnt 0" ::: "memory");
}

// ---------------------------------------------------------------------------
extern "C" void kernel_launch(void* const* d_in, const int* in_sizes, int n_in,
                              void* d_out, int out_size, void* d_ws, size_t ws_size,
                              hipStream_t stream) {
    const float4* cls4   = (const float4*)d_in[0];   // (1,18,480,480) f32
    const float*  deltas = (const float*)d_in[1];    // (1,36,480,480) f32
    float* out = (float*)d_out;                      // 1500 floats

    uint32_t* ws = (uint32_t*)d_ws;
    uint32_t* hist    = ws + 0;          // 4096
    int*      binB    = (int*)(ws + 4096);
    uint32_t* counter = ws + 4097;
    uint32_t* ckey    = ws + 4100;       // CAP
    uint32_t* cidx    = ws + 4100 + CAP; // CAP
    float*    sscore  = (float*)(ws + 4100 + 2 * CAP);          // 3000
    uint32_t* ssidx   = ws + 4100 + 2 * CAP + NPRE;             // 3000
    float4*   boxes4  = (float4*)(ws + 4100 + 2 * CAP + 2 * NPRE);        // 3000*4, 16B aligned
    uint32_t* maskbuf = ws + 4100 + 2 * CAP + 2 * NPRE + 4 * NPRE;        // 3000*128

    k0_init   <<<32, 256, 0, stream>>>(hist, counter, sscore, ssidx, out);
    k1_hist   <<<256, 256, 0, stream>>>(cls4, hist);
    k2_thresh <<<1, 32, 0, stream>>>(hist, binB);
    k3_compact<<<256, 256, 0, stream>>>(cls4, binB, counter, ckey, cidx);
    k4_rank   <<<CAP / 256, 256, 0, stream>>>(ckey, cidx, counter, sscore, ssidx);
    k5_boxes  <<<(NPRE + 255) / 256, 256, 0, stream>>>(ssidx, deltas, boxes4);
    k6_mask   <<<(NPRE * 94 + 255) / 256, 256, 0, stream>>>(boxes4, maskbuf);
    k7_scan   <<<1, 32, 0, stream>>>(maskbuf, sscore, boxes4, out);
}